// SGC_33208687133422
// MI455X (gfx1250) — compile-verified
//
#include <hip/hip_runtime.h>
#include <hip/hip_bf16.h>
#include <math.h>

// Problem constants (match reference)
#define NN   50000
#define EE   800000
#define CC   5000
#define DD   128
#define OUTF 40
#define NPAD 48           // OUT padded to 3 WMMA n-tiles of 16
#define CPAD 5008         // C padded to multiple of 16 (313 tiles)

typedef __attribute__((ext_vector_type(2))) float v2f;
typedef __attribute__((ext_vector_type(8))) float v8f;

// ---------------- degree / dinv ----------------
__global__ void deg_init_kernel(float* __restrict__ deg) {
    int i = blockIdx.x * blockDim.x + threadIdx.x;
    if (i < NN) deg[i] = 1.0f;                      // self-loop
}

__global__ void deg_edge_kernel(float* __restrict__ deg, const int* __restrict__ ei) {
    int e = blockIdx.x * blockDim.x + threadIdx.x;
    if (e >= EE) return;
    int d = ei[EE + e];                              // dst row
    if (d < 0) d = 0; if (d >= NN) d = NN - 1;
    unsafeAtomicAdd(deg + d, 1.0f);                  // hardware global_atomic_add_f32
}

__global__ void dinv_kernel(float* __restrict__ deg) {
    int i = blockIdx.x * blockDim.x + threadIdx.x;
    if (i < NN) deg[i] = rsqrtf(deg[i]);             // deg >= 1 always
}

// ---------------- propagation ----------------
// h_out[i,:] = dinv[i]^2 * h_in[i,:]   (self-loop term; also initializes h_out)
__global__ void self_init_kernel(const float* __restrict__ hin,
                                 float* __restrict__ hout,
                                 const float* __restrict__ dinv) {
    int t = blockIdx.x * blockDim.x + threadIdx.x;   // N*32 threads, float4 each
    if (t >= NN * 32) return;
    int i = t >> 5;
    int d = (t & 31) * 4;
    float di = dinv[i];
    float c = di * di;
    const float4 v = *(const float4*)(hin + (size_t)i * DD + d);
    float4 o;
    o.x = c * v.x; o.y = c * v.y; o.z = c * v.z; o.w = c * v.w;
    *(float4*)(hout + (size_t)i * DD + d) = o;
}

// h_out[dst,:] += dinv[src]*dinv[dst] * h_in[src,:]   (one 32-lane group per edge)
__global__ void prop_edge_kernel(const float* __restrict__ hin,
                                 float* __restrict__ hout,
                                 const float* __restrict__ dinv,
                                 const int* __restrict__ ei) {
    int t = blockIdx.x * blockDim.x + threadIdx.x;   // E*32 threads
    int e = t >> 5;
    if (e >= EE) return;
    int d = (t & 31) * 4;
    int s  = ei[e];
    int dd = ei[EE + e];
    if (s  < 0) s  = 0; if (s  >= NN) s  = NN - 1;
    if (dd < 0) dd = 0; if (dd >= NN) dd = NN - 1;
    float c = dinv[s] * dinv[dd];
    const float4 v = *(const float4*)(hin + (size_t)s * DD + d);
    float* o = hout + (size_t)dd * DD + d;
    unsafeAtomicAdd(o + 0, c * v.x);
    unsafeAtomicAdd(o + 1, c * v.y);
    unsafeAtomicAdd(o + 2, c * v.z);
    unsafeAtomicAdd(o + 3, c * v.w);
}

// ---------------- compressed linear: yc[c, n] = sum_k h[rep[c], k] * W[n, k] + b[n]
// V_WMMA_F32_16X16X4_F32, one wave per 16x48 tile; 4 waves per block.
__global__ __launch_bounds__(128)
void gemm_wmma_kernel(const float* __restrict__ h,
                      const int* __restrict__ rep_idx,
                      const float* __restrict__ W,
                      const float* __restrict__ bias,
                      float* __restrict__ yc) {
    const int lane = threadIdx.x & 31;
    const int wave = threadIdx.x >> 5;
    const int tile = blockIdx.x * 4 + wave;          // 16 clusters per tile
    const int ntiles = CPAD / 16;                    // 313
    if (tile >= ntiles) return;                      // wave-uniform exit

    const int m  = lane & 15;                        // A row / B col within tile
    const int kh = (lane >> 4) * 2;                  // K sub-offset {0,2}

    // A row pointer: representative node row for cluster (tile*16 + m)
    int row = tile * 16 + m;
    int rr  = rep_idx[row < CC ? row : (CC - 1)];
    if (rr < 0) rr = 0; if (rr >= NN) rr = NN - 1;
    const float* arow = h + (size_t)rr * DD;

    // B column pointers for the 3 n-tiles (W is [OUT, D] row-major; B[k][n] = W[n][k])
    const int n0 = 0  + m, n1 = 16 + m, n2 = 32 + m;
    const bool v2ok = (n2 < OUTF);                   // n0,n1 always < 40
    const float* w0 = W + (size_t)n0 * DD;
    const float* w1 = W + (size_t)n1 * DD;
    const float* w2 = W + (size_t)(v2ok ? n2 : 0) * DD;

    v8f acc0 = {}; v8f acc1 = {}; v8f acc2 = {};

#pragma unroll 4
    for (int k = 0; k < DD; k += 4) {
        const float2 av = *(const float2*)(arow + k + kh);
        v2f a; a.x = av.x; a.y = av.y;

        const float2 b0 = *(const float2*)(w0 + k + kh);
        const float2 b1 = *(const float2*)(w1 + k + kh);
        float2 b2 = *(const float2*)(w2 + k + kh);
        if (!v2ok) { b2.x = 0.0f; b2.y = 0.0f; }

        v2f bb0; bb0.x = b0.x; bb0.y = b0.y;
        v2f bb1; bb1.x = b1.x; bb1.y = b1.y;
        v2f bb2; bb2.x = b2.x; bb2.y = b2.y;

        acc0 = __builtin_amdgcn_wmma_f32_16x16x4_f32(false, a, false, bb0, (short)0, acc0, false, false);
        acc1 = __builtin_amdgcn_wmma_f32_16x16x4_f32(false, a, false, bb1, (short)0, acc1, false, false);
        acc2 = __builtin_amdgcn_wmma_f32_16x16x4_f32(false, a, false, bb2, (short)0, acc2, false, false);
    }

    // bias per output column (column index is lane-constant: n = nt*16 + m)
    const float bz0 = bias[n0];
    const float bz1 = bias[n1];
    const float bz2 = v2ok ? bias[n2] : 0.0f;

    // C/D layout: lane l, VGPR v -> row (v + 8*(l/16)), col l%16
    const int rbase = tile * 16 + 8 * (lane >> 4);
#pragma unroll
    for (int v = 0; v < 8; ++v) {
        int rg = rbase + v;
        if (rg < CC) {
            float* orow = yc + (size_t)rg * NPAD;
            orow[n0] = acc0[v] + bz0;
            orow[n1] = acc1[v] + bz1;
            orow[n2] = acc2[v] + bz2;
        }
    }
}

// ---------------- per-cluster log_softmax over first 40 columns ----------------
__global__ void logsoftmax_kernel(const float* __restrict__ yc, float* __restrict__ lsm) {
    int c = blockIdx.x * blockDim.x + threadIdx.x;
    if (c >= CC) return;
    const float* r = yc + (size_t)c * NPAD;
    float mx = -3.402823466e38f;
#pragma unroll
    for (int o = 0; o < OUTF; ++o) mx = fmaxf(mx, r[o]);
    float s = 0.0f;
#pragma unroll
    for (int o = 0; o < OUTF; ++o) s += __expf(r[o] - mx);
    float l = __logf(s);
    float* w = lsm + (size_t)c * OUTF;
#pragma unroll
    for (int o = 0; o < OUTF; ++o) w[o] = r[o] - mx - l;
}

// ---------------- scatter lsm[cluster[i]] -> out[i]  (float4 copy) ----------------
__global__ void scatter_kernel(const float* __restrict__ lsm,
                               const int* __restrict__ cl,
                               float* __restrict__ out) {
    int t = blockIdx.x * blockDim.x + threadIdx.x;   // N * 10 float4's
    if (t >= NN * 10) return;
    int i = t / 10;
    int q = t % 10;
    int c = cl[i];
    if (c < 0) c = 0; if (c >= CC) c = CC - 1;
    const float4 v = ((const float4*)(lsm + (size_t)c * OUTF))[q];
    ((float4*)(out + (size_t)i * OUTF))[q] = v;
}

extern "C" void kernel_launch(void* const* d_in, const int* in_sizes, int n_in,
                              void* d_out, int out_size, void* d_ws, size_t ws_size,
                              hipStream_t stream) {
    const float* x   = (const float*)d_in[0];        // [N, D]
    const int*   ei  = (const int*)  d_in[1];        // [2, E]
    const int*   cl  = (const int*)  d_in[2];        // [N]
    const int*   rep = (const int*)  d_in[3];        // [C]
    const float* W   = (const float*)d_in[4];        // [OUT, D]
    const float* b   = (const float*)d_in[5];        // [OUT]
    float* out = (float*)d_out;                      // [N, OUT]

    float* ws   = (float*)d_ws;
    float* dinv = ws;                                // N        (reused: deg -> dinv)
    float* h1   = dinv + NN;                         // N*D
    float* h2   = h1 + (size_t)NN * DD;              // N*D
    float* yc   = h2 + (size_t)NN * DD;              // CPAD*48
    float* lsm  = yc + (size_t)CPAD * NPAD;          // C*40

    const int B = 256;

    // degree -> dinv
    deg_init_kernel<<<(NN + B - 1) / B, B, 0, stream>>>(dinv);
    deg_edge_kernel<<<(EE + B - 1) / B, B, 0, stream>>>(dinv, ei);
    dinv_kernel<<<(NN + B - 1) / B, B, 0, stream>>>(dinv);

    // hop 1: h1 = D^-1/2 (A+I) D^-1/2 x
    self_init_kernel<<<(NN * 32 + B - 1) / B, B, 0, stream>>>(x, h1, dinv);
    prop_edge_kernel<<<(EE * 32 + B - 1) / B, B, 0, stream>>>(x, h1, dinv, ei);

    // hop 2: h2 = D^-1/2 (A+I) D^-1/2 h1
    self_init_kernel<<<(NN * 32 + B - 1) / B, B, 0, stream>>>(h1, h2, dinv);
    prop_edge_kernel<<<(EE * 32 + B - 1) / B, B, 0, stream>>>(h1, h2, dinv, ei);

    // compressed linear on representative rows (WMMA f32 16x16x4)
    gemm_wmma_kernel<<<(CPAD / 16 + 3) / 4, 128, 0, stream>>>(h2, rep, W, b, yc);

    // per-cluster log_softmax, then scatter to all nodes
    logsoftmax_kernel<<<(CC + B - 1) / B, B, 0, stream>>>(yc, lsm);
    scatter_kernel<<<(NN * 10 + B - 1) / B, B, 0, stream>>>(lsm, cl, out);
}